// Reservoir_60387240182015
// MI455X (gfx1250) — compile-verified
//
#include <hip/hip_runtime.h>
#include <math.h>

#define N_RES   2048
#define N_IN    64
#define N_OUT   16
#define T_STEPS 4096
#define K_TOT   (N_IN + N_OUT)      // 80
#define NBLK    64                  // persistent blocks; whole W fits in 64 * 256KB LDS
#define ROWS_PER_WG (N_RES / NBLK)  // 32
#define TPR     8                   // threads per row in the dot product
#define BLOCK   256
#define W_STRIDE 2056               // 2048 + 8 pad -> bank = (8*row + sub + 8k)%64, conflict-free

typedef __attribute__((ext_vector_type(2))) float v2f;
typedef __attribute__((ext_vector_type(8))) float v8f;

// ---------------------------------------------------------------------------
// init: zero grid-barrier counter and state ping-pong buffers (ws is poisoned)
// ---------------------------------------------------------------------------
__global__ void esn_init(unsigned* counter, float* buf0, float* buf1) {
  int tid = threadIdx.x;
  if (tid < 64) counter[tid] = 0u;
  for (int i = tid; i < N_RES; i += BLOCK) { buf0[i] = 0.f; buf1[i] = 0.f; }
}

// ---------------------------------------------------------------------------
// pre[t][r] = sum_j Wc[r][j] * X[t][j],  X = [inputs[t], outputs[t-1]], K=80
// exact-f32 WMMA GEMM: one wave per 16(t) x 16(r) tile, K in chunks of 4.
// A 16x4 layout: lane -> M=lane%16, VGPR v + half h -> K = 2h+v
// B 4x16 layout: lane -> N=lane%16, VGPR v + half h -> K = 2h+v
// D 16x16:       lane -> N=lane%16, VGPR i + half h -> M = i + 8h
// Out-of-range time rows are CLAMPED (not zeroed): a stale A-row only
// pollutes its own D-row, which is masked at the store. This keeps the
// loads unconditional (no EXEC manipulation) and b64-vectorizable.
// ---------------------------------------------------------------------------
__global__ __launch_bounds__(BLOCK) void esn_pre_wmma(
    const float* __restrict__ inputs, const float* __restrict__ outputs,
    const float* __restrict__ w_in, const float* __restrict__ w_fb,
    float* __restrict__ pre) {
  int wave   = blockIdx.x * (BLOCK >> 5) + (threadIdx.x >> 5);
  int lane   = threadIdx.x & 31;
  int tile_m = wave & 255;   // 256 t-tiles (4095 steps, last tile partial)
  int tile_n = wave >> 8;    // 128 r-tiles
  int lo = lane & 15;
  int hi = lane >> 4;

  int ti = tile_m * 16 + lo;                           // 0..4095
  int tl = (ti < T_STEPS - 1 ? ti : T_STEPS - 2) + 1;  // clamped time 1..4095
  int r  = tile_n * 16 + lo;

  v8f acc = {};
#pragma unroll
  for (int k0 = 0; k0 < K_TOT; k0 += 4) {
    int j = k0 + 2 * hi;  // pair (j, j+1) never straddles the 64-boundary
    v2f a = (k0 < N_IN)
                ? *(const v2f*)&inputs[tl * N_IN + j]
                : *(const v2f*)&outputs[(tl - 1) * N_OUT + (j - N_IN)];
    v2f b = (k0 < N_IN)
                ? *(const v2f*)&w_in[r * N_IN + j]
                : *(const v2f*)&w_fb[r * N_OUT + (j - N_IN)];
    acc = __builtin_amdgcn_wmma_f32_16x16x4_f32(
        /*neg_a=*/false, a, /*neg_b=*/false, b,
        /*c_mod=*/(short)0, acc, /*reuse_a=*/false, /*reuse_b=*/false);
  }

#pragma unroll
  for (int i = 0; i < 8; ++i) {
    int tm = tile_m * 16 + (i + 8 * hi);
    if (tm < (T_STEPS - 1))
      pre[(size_t)(tm + 1) * N_RES + tile_n * 16 + lo] = acc[i];
  }
}

// ---------------------------------------------------------------------------
// persistent recurrence: 64 blocks, each pins 32 rows of W in LDS (257 KB),
// one grid barrier per step, double-buffered state in global memory.
// ---------------------------------------------------------------------------
__global__ __launch_bounds__(BLOCK, 1) void esn_recur(
    const float* __restrict__ w, const float* __restrict__ inputs,
    const float* __restrict__ outputs, const float* __restrict__ noise,
    const float* __restrict__ w_in, const float* __restrict__ w_fb,
    const float* __restrict__ pre, float* buf0, float* buf1,
    unsigned* counter, int use_pre) {
  __shared__ float w_lds[ROWS_PER_WG * W_STRIDE];  // 263168 B
  __shared__ float s_lds[N_RES];                   //   8192 B
  __shared__ float red[BLOCK];                     //   1024 B

  const int tid   = threadIdx.x;
  const int row   = tid >> 3;   // 0..31
  const int sub   = tid & 7;    // 0..7
  const int rbase = blockIdx.x * ROWS_PER_WG;

  // stage this block's W rows into LDS (coalesced, once)
  for (int idx = tid; idx < ROWS_PER_WG * N_RES; idx += BLOCK) {
    int rr = idx >> 11;               // / 2048
    int cc = idx & (N_RES - 1);
    w_lds[rr * W_STRIDE + cc] = w[(size_t)(rbase + rr) * N_RES + cc];
  }

  const float* cur = buf0;
  float*       nxt = buf1;

  for (int s = 1; s < T_STEPS; ++s) {
    // broadcast current state into LDS
    for (int i = tid; i < N_RES; i += BLOCK) s_lds[i] = cur[i];
    __syncthreads();

    // partial dot: 8 threads per row, bank-conflict-free stride-8 walk
    const float* wr = &w_lds[row * W_STRIDE + sub];
    float acc = 0.f;
#pragma unroll 8
    for (int k = 0; k < N_RES / TPR; ++k)
      acc = fmaf(wr[8 * k], s_lds[sub + 8 * k], acc);

    if (!use_pre) {
      // fallback: fold W_in*inp + W_fb*outp into the same reduction
      int rg = rbase + row;
#pragma unroll
      for (int q = 0; q < K_TOT / TPR; ++q) {
        int j = sub + 8 * q;  // covers 0..79 exactly once across subs
        float xv = (j < N_IN) ? inputs[s * N_IN + j]
                              : outputs[(s - 1) * N_OUT + (j - N_IN)];
        float wv = (j < N_IN) ? w_in[rg * N_IN + j]
                              : w_fb[rg * N_OUT + (j - N_IN)];
        acc = fmaf(wv, xv, acc);
      }
    }

    red[tid] = acc;
    __syncthreads();

    if (tid < ROWS_PER_WG) {
      float sum = 0.f;
#pragma unroll
      for (int q = 0; q < TPR; ++q) sum += red[tid * TPR + q];
      int rg = rbase + tid;
      if (use_pre) sum += pre[(size_t)s * N_RES + rg];
      nxt[rg] = tanhf(sum) + noise[(size_t)s * N_RES + rg];
    }
    __syncthreads();

    // grid barrier: monotonic counter, release->add->spin->acquire
    if (tid == 0) {
      __threadfence();
      atomicAdd(counter, 1u);
      unsigned target = (unsigned)s * NBLK;
      while (atomicAdd(counter, 0u) < target) __builtin_amdgcn_s_sleep(1);
      __threadfence();
    }
    __syncthreads();

    float* tmp = (float*)cur; cur = nxt; nxt = tmp;
  }
}

// ---------------------------------------------------------------------------
// finalize: out[0:16] = last @ w_out ; out[16:2144] = concat(last, in[-1], out[-2])
// ---------------------------------------------------------------------------
__global__ void esn_final(const float* __restrict__ last,
                          const float* __restrict__ w_out,
                          const float* __restrict__ inputs,
                          const float* __restrict__ outputs,
                          float* __restrict__ out) {
  __shared__ float partial[BLOCK];
  int tid = threadIdx.x;
  int o = tid & 15, chunk = tid >> 4;  // 16 chunks of 128 rows per output
  float s = 0.f;
  for (int r = chunk * 128; r < chunk * 128 + 128; ++r)
    s = fmaf(last[r], w_out[r * N_OUT + o], s);
  partial[tid] = s;
  __syncthreads();
  if (tid < N_OUT) {
    float t = 0.f;
#pragma unroll
    for (int c = 0; c < 16; ++c) t += partial[c * 16 + tid];
    out[tid] = t;
  }
  for (int i = tid; i < N_RES; i += BLOCK) out[N_OUT + i] = last[i];
  if (tid < N_IN)  out[N_OUT + N_RES + tid] = inputs[(T_STEPS - 1) * N_IN + tid];
  if (tid < N_OUT) out[N_OUT + N_RES + N_IN + tid] = outputs[(T_STEPS - 2) * N_OUT + tid];
}

// ---------------------------------------------------------------------------
extern "C" void kernel_launch(void* const* d_in, const int* in_sizes, int n_in,
                              void* d_out, int out_size, void* d_ws, size_t ws_size,
                              hipStream_t stream) {
  (void)in_sizes; (void)n_in; (void)out_size;
  const float* inputs  = (const float*)d_in[0];
  const float* outputs = (const float*)d_in[1];
  const float* noise   = (const float*)d_in[2];
  const float* w       = (const float*)d_in[3];
  const float* w_in    = (const float*)d_in[4];
  const float* w_fb    = (const float*)d_in[5];
  const float* w_out   = (const float*)d_in[6];
  float* out = (float*)d_out;

  char*     ws      = (char*)d_ws;
  unsigned* counter = (unsigned*)ws;                         // 256 B reserved
  float*    buf0    = (float*)(ws + 256);
  float*    buf1    = (float*)(ws + 256 + N_RES * 4);
  float*    pre     = (float*)(ws + 256 + 2 * N_RES * 4);
  size_t need = 256 + (size_t)2 * N_RES * 4 + (size_t)T_STEPS * N_RES * 4;
  int use_pre = (ws_size >= need) ? 1 : 0;                   // deterministic per run

  esn_init<<<1, BLOCK, 0, stream>>>(counter, buf0, buf1);
  if (use_pre)
    esn_pre_wmma<<<(256 * 128) / (BLOCK / 32), BLOCK, 0, stream>>>(
        inputs, outputs, w_in, w_fb, pre);
  esn_recur<<<NBLK, BLOCK, 0, stream>>>(w, inputs, outputs, noise, w_in, w_fb,
                                        pre, buf0, buf1, counter, use_pre);
  // 4095 steps (odd) -> last state lives in buf1
  esn_final<<<1, BLOCK, 0, stream>>>(buf1, w_out, inputs, outputs, out);
}